// _EdgesToNodesAggregator_83631603188041
// MI455X (gfx1250) — compile-verified
//
#include <hip/hip_runtime.h>
#include <hip/hip_bf16.h>

// segment_sum: out[r] += edges[e] for r = receivers[e]
//   edges:     [625000, 128] f32  (320 MB, streamed once -> non-temporal loads)
//   receivers: [625000]      i32  (2.5 MB, wave-uniform scalar loads)
//   out:       [50000, 128]  f32  (25.6 MB, L2-resident accumulator via f32 atomics)
//
// Bandwidth-bound (~350 MB total, ~15 us floor @ 23.3 TB/s). No matrix math:
// WMMA does not apply. CDNA5-specific paths used instead:
//   * global_load_b128 with TH=NT for the 320 MB edge stream (don't thrash the
//     192 MB L2 where the 25.6 MB accumulator lives)
//   * global_atomic_add_f32 no-return (STOREcnt-tracked, no RAW stalls)
//   * global_prefetch_b8 for the next grid-stride row
//   * s_load_b32 for the wave-uniform receiver index (readfirstlane-forced)
//   * wave32 mapping: one wave == one 128-float edge row (32 lanes x float4)

typedef float v4f __attribute__((ext_vector_type(4)));

#define D_FEAT 128

__global__ void _seg_zero_kernel(v4f* __restrict__ out, int n4) {
    int i = blockIdx.x * blockDim.x + threadIdx.x;
    const int stride = gridDim.x * blockDim.x;
    const v4f z = (v4f)0.0f;
    for (; i < n4; i += stride) {
        out[i] = z;
    }
}

__global__ void _seg_scatter_kernel(const float* __restrict__ edges,
                                    const int* __restrict__ recv,
                                    float* __restrict__ out,
                                    int n_edges) {
    const int lane   = threadIdx.x & 31;                                  // wave32
    const int wave0  = (blockIdx.x * blockDim.x + threadIdx.x) >> 5;
    const int nwaves = (gridDim.x * blockDim.x) >> 5;

    // Force wave-uniformity so index math + receiver load go scalar (SALU/SMEM).
    const int wave = __builtin_amdgcn_readfirstlane(wave0);

    for (int e = wave; e < n_edges; e += nwaves) {
        // Prefetch next iteration's row chunk (gfx1250 global_prefetch_b8).
        const int en = e + nwaves;
        if (en < n_edges) {
            __builtin_prefetch(edges + (size_t)en * D_FEAT + lane * 4, 0, 0);
        }

        // Streaming (non-temporal) 128-bit load: 32 lanes x float4 = 512 B row.
        const v4f* row = (const v4f*)(edges + (size_t)e * D_FEAT);
        const v4f v = __builtin_nontemporal_load(row + lane);

        // Wave-uniform receiver -> scalar load through constant cache.
        const int r = recv[e];
        float* dst = out + (size_t)r * D_FEAT + lane * 4;

        // No-return f32 atomic adds; RMW executes at the L2 atomic units.
        unsafeAtomicAdd(dst + 0, v.x);
        unsafeAtomicAdd(dst + 1, v.y);
        unsafeAtomicAdd(dst + 2, v.z);
        unsafeAtomicAdd(dst + 3, v.w);
    }
}

extern "C" void kernel_launch(void* const* d_in, const int* in_sizes, int n_in,
                              void* d_out, int out_size, void* d_ws, size_t ws_size,
                              hipStream_t stream) {
    const float* edges = (const float*)d_in[0];
    const int*   recv  = (const int*)d_in[1];
    // d_in[2] is num_nodes (scalar) — out_size already equals num_nodes * D_FEAT.
    float* out = (float*)d_out;

    const int n_edges = in_sizes[0] / D_FEAT;   // 625000
    const int n4      = out_size / 4;           // out as float4 count

    const int threads = 256;

    // 1) Zero the accumulator (harness poisons d_out; atomics need zeros each call).
    _seg_zero_kernel<<<4096, threads, 0, stream>>>((v4f*)out, n4);

    // 2) Wave-per-row atomic scatter. 4096 blocks * 8 waves = 32768 waves
    //    -> ~19 edge rows per wave, deep enough to hide HBM + atomic latency.
    _seg_scatter_kernel<<<4096, threads, 0, stream>>>(edges, recv, out, n_edges);
}